// AttentionPooling_41094247088907
// MI455X (gfx1250) — compile-verified
//
#include <hip/hip_runtime.h>
#include <math.h>

// ---- problem constants (from reference) ----
#define BATCH     32
#define SEQ       784
#define DIM       1024
#define NLAYERS   3
#define D4        256          // DIM/4
#define NCLASS    400
#define KCLS      768          // NLAYERS * D4
#define FEAT_BSTRIDE (SEQ * DIM)   // 802816
#define SCHUNK    4
#define SLEN      (SEQ / SCHUNK)   // 196

typedef __attribute__((ext_vector_type(2))) float v2f;
typedef __attribute__((ext_vector_type(8))) float v8f;

// ------------------------------------------------------------------
// Kernel 1: scores[b,l,s] = sum_d (feat[b,s,d]+pe[s,d]) * attn_w[l,d]
// One wave per (b,s); each lane loads float4 (b128) - 8 iterations.
// grid = 32*784/8 blocks of 256 threads (8 waves/block).
// ------------------------------------------------------------------
__global__ void scores_kernel(const float* __restrict__ feat,
                              const float* __restrict__ pe,
                              const float* __restrict__ attn_w,
                              float* __restrict__ scores) {
  int wave = (blockIdx.x << 3) + (threadIdx.x >> 5);   // 0 .. 25087
  int lane = threadIdx.x & 31;
  int b = wave / SEQ;
  int s = wave - b * SEQ;

  const float4* f  = (const float4*)(feat + (size_t)b * FEAT_BSTRIDE
                                          + (size_t)s * DIM) + lane;
  const float4* p  = (const float4*)(pe + (size_t)s * DIM) + lane;
  const float4* w0 = (const float4*)(attn_w)           + lane;
  const float4* w1 = (const float4*)(attn_w + DIM)     + lane;
  const float4* w2 = (const float4*)(attn_w + 2 * DIM) + lane;

  float a0 = 0.f, a1 = 0.f, a2 = 0.f;
  #pragma unroll
  for (int j = 0; j < DIM / (32 * 4); ++j) {           // 8 iterations
    float4 x  = f[j * 32];
    float4 pp = p[j * 32];
    x.x += pp.x; x.y += pp.y; x.z += pp.z; x.w += pp.w;
    float4 q0 = w0[j * 32], q1 = w1[j * 32], q2 = w2[j * 32];
    a0 = fmaf(x.x, q0.x, fmaf(x.y, q0.y, fmaf(x.z, q0.z, fmaf(x.w, q0.w, a0))));
    a1 = fmaf(x.x, q1.x, fmaf(x.y, q1.y, fmaf(x.z, q1.z, fmaf(x.w, q1.w, a1))));
    a2 = fmaf(x.x, q2.x, fmaf(x.y, q2.y, fmaf(x.z, q2.z, fmaf(x.w, q2.w, a2))));
  }
  // wave32 reduction
  for (int off = 16; off > 0; off >>= 1) {
    a0 += __shfl_xor(a0, off);
    a1 += __shfl_xor(a1, off);
    a2 += __shfl_xor(a2, off);
  }
  if (lane == 0) {
    float* sc = scores + (size_t)b * NLAYERS * SEQ + s;
    sc[0]       = a0;
    sc[SEQ]     = a1;
    sc[2 * SEQ] = a2;
  }
}

// ------------------------------------------------------------------
// Kernel 2: in-place softmax over S for each of B*L = 96 rows.
// ------------------------------------------------------------------
__global__ void softmax_kernel(float* __restrict__ sc) {
  __shared__ float red[256];
  int tid = threadIdx.x;
  float* p = sc + (size_t)blockIdx.x * SEQ;

  float m = -INFINITY;
  for (int i = tid; i < SEQ; i += 256) m = fmaxf(m, p[i]);
  red[tid] = m; __syncthreads();
  for (int s = 128; s > 0; s >>= 1) {
    if (tid < s) red[tid] = fmaxf(red[tid], red[tid + s]);
    __syncthreads();
  }
  m = red[0]; __syncthreads();

  float sum = 0.f;
  for (int i = tid; i < SEQ; i += 256) {
    float e = __expf(p[i] - m);
    p[i] = e;
    sum += e;
  }
  red[tid] = sum; __syncthreads();
  for (int s = 128; s > 0; s >>= 1) {
    if (tid < s) red[tid] += red[tid + s];
    __syncthreads();
  }
  float inv = 1.f / red[0];
  __syncthreads();
  for (int i = tid; i < SEQ; i += 256) p[i] *= inv;
}

// ------------------------------------------------------------------
// Kernel 3a: partial attended over an S-chunk.
// part[b,c,l,d] = sum_{s in chunk c} w[b,l,s] * (feat[b,s,d]+pe[s,d])
// Block = 256 threads, each owns 4 consecutive d (float4) -> covers D=1024.
// grid = 32 b * 4 chunks = 128 blocks (1024 waves). Second feature sweep
// is served from L2 (103MB < 192MB).
// ------------------------------------------------------------------
__global__ void attend_part_kernel(const float* __restrict__ feat,
                                   const float* __restrict__ pe,
                                   const float* __restrict__ w,
                                   float* __restrict__ part) {
  __shared__ float wl[NLAYERS * SLEN];      // 588 floats
  int b  = blockIdx.x >> 2;
  int c  = blockIdx.x & 3;
  int s0 = c * SLEN;
  int tid = threadIdx.x;

  for (int i = tid; i < NLAYERS * SLEN; i += 256) {
    int l = i / SLEN, j = i - l * SLEN;
    wl[i] = w[(size_t)b * NLAYERS * SEQ + (size_t)l * SEQ + s0 + j];
  }
  __syncthreads();

  const float4* f = (const float4*)(feat + (size_t)b * FEAT_BSTRIDE
                                         + (size_t)s0 * DIM) + tid;
  const float4* p = (const float4*)(pe + (size_t)s0 * DIM) + tid;

  float4 a0 = {0.f, 0.f, 0.f, 0.f};
  float4 a1 = {0.f, 0.f, 0.f, 0.f};
  float4 a2 = {0.f, 0.f, 0.f, 0.f};
  for (int s = 0; s < SLEN; ++s) {
    float4 x  = f[s * (DIM / 4)];
    float4 pp = p[s * (DIM / 4)];
    x.x += pp.x; x.y += pp.y; x.z += pp.z; x.w += pp.w;
    float u0 = wl[s], u1 = wl[SLEN + s], u2 = wl[2 * SLEN + s];
    a0.x = fmaf(u0, x.x, a0.x); a0.y = fmaf(u0, x.y, a0.y);
    a0.z = fmaf(u0, x.z, a0.z); a0.w = fmaf(u0, x.w, a0.w);
    a1.x = fmaf(u1, x.x, a1.x); a1.y = fmaf(u1, x.y, a1.y);
    a1.z = fmaf(u1, x.z, a1.z); a1.w = fmaf(u1, x.w, a1.w);
    a2.x = fmaf(u2, x.x, a2.x); a2.y = fmaf(u2, x.y, a2.y);
    a2.z = fmaf(u2, x.z, a2.z); a2.w = fmaf(u2, x.w, a2.w);
  }
  float4* o = (float4*)(part + (((size_t)b * SCHUNK + c) * NLAYERS) * DIM) + tid;
  o[0]               = a0;
  o[DIM / 4]         = a1;
  o[2 * (DIM / 4)]   = a2;
}

// Kernel 3b: attended[b,l,d] = sum_c part[b,c,l,d]  (deterministic reduce)
__global__ void attend_reduce_kernel(const float* __restrict__ part,
                                     float* __restrict__ att) {
  int i  = blockIdx.x * 256 + threadIdx.x;   // over 96*1024
  int bl = i >> 10;                          // b*3 + l
  int d  = i & 1023;
  int b  = bl / NLAYERS;
  int l  = bl - b * NLAYERS;
  float sum = 0.f;
  #pragma unroll
  for (int c = 0; c < SCHUNK; ++c)
    sum += part[(((size_t)b * SCHUNK + c) * NLAYERS + l) * DIM + d];
  att[i] = sum;
}

// ------------------------------------------------------------------
// WMMA f32 GEMM tile worker: one wave computes one 16x16 tile of
// C = act( A(MxK) * W(KxN) + bias ).  V_WMMA_F32_16X16X4_F32.
// A layout: lane<16 -> K=k,k+1 ; lane>=16 -> K=k+2,k+3  (row m0+lane&15)
// B layout: VGPR0 -> K=k (lo)/k+2 (hi), VGPR1 -> K=k+1/k+3, col n0+lane&15
// ------------------------------------------------------------------
template <int K, int N, bool RELU>
__device__ __forceinline__ void gemm_tile_f32(const float* __restrict__ A,
                                              const float* __restrict__ W,
                                              const float* __restrict__ bias,
                                              float* __restrict__ out,
                                              int m0, int n0) {
  int lane = threadIdx.x & 31;
  int half = lane >> 4;        // 0: K pair {k,k+1}; 1: {k+2,k+3}
  int l16  = lane & 15;

  const float* arow = A + (size_t)(m0 + l16) * K + (half << 1);
  const float* bcol = W + (size_t)(half << 1) * N + n0 + l16;

  v8f c = {};
  for (int k = 0; k < K; k += 4) {
    v2f a, bb;
    a.x  = arow[k];
    a.y  = arow[k + 1];
    bb.x = bcol[(size_t)k * N];
    bb.y = bcol[(size_t)(k + 1) * N];
    // 8 args: (neg_a, A, neg_b, B, c_mod, C, reuse_a, reuse_b)
    c = __builtin_amdgcn_wmma_f32_16x16x4_f32(false, a, false, bb,
                                              (short)0, c, false, false);
  }
  float bv = bias[n0 + l16];
  #pragma unroll
  for (int r = 0; r < 8; ++r) {
    int row = m0 + r + (half << 3);
    float v = c[r] + bv;
    if (RELU) v = v > 0.f ? v : 0.f;
    out[(size_t)row * N + n0 + l16] = v;
  }
}

// enc = relu(attended(96x1024) @ enc_W(1024x256) + enc_b) : 6*16 tiles
__global__ void enc_gemm_kernel(const float* __restrict__ A,
                                const float* __restrict__ W,
                                const float* __restrict__ bias,
                                float* __restrict__ out) {
  int tile = blockIdx.x;                  // 96 tiles
  int m0 = (tile >> 4) << 4;              // 6 M-tiles
  int n0 = (tile & 15) << 4;              // 16 N-tiles
  gemm_tile_f32<DIM, D4, true>(A, W, bias, out, m0, n0);
}

// out = enc(32x768) @ cls_W(768x400) + cls_b : 2*25 tiles
__global__ void cls_gemm_kernel(const float* __restrict__ A,
                                const float* __restrict__ W,
                                const float* __restrict__ bias,
                                float* __restrict__ out) {
  int tile = blockIdx.x;                  // 50 tiles
  int mt = tile / 25;
  int nt = tile - mt * 25;
  gemm_tile_f32<KCLS, NCLASS, false>(A, W, bias, out, mt << 4, nt << 4);
}

extern "C" void kernel_launch(void* const* d_in, const int* in_sizes, int n_in,
                              void* d_out, int out_size, void* d_ws, size_t ws_size,
                              hipStream_t stream) {
  const float* feat   = (const float*)d_in[0];  // 512 x 50176
  const float* pe     = (const float*)d_in[1];  // 784 x 1024
  const float* attn_w = (const float*)d_in[2];  // 3 x 1024
  const float* enc_W  = (const float*)d_in[3];  // 1024 x 256
  const float* enc_b  = (const float*)d_in[4];  // 256
  const float* cls_W  = (const float*)d_in[5];  // 768 x 400
  const float* cls_b  = (const float*)d_in[6];  // 400
  float* out = (float*)d_out;                   // 32 x 400

  float* ws       = (float*)d_ws;
  float* scores   = ws;                                   // 32*3*784  = 75264
  float* attended = scores + BATCH * NLAYERS * SEQ;       // 96*1024   = 98304
  float* enc      = attended + BATCH * NLAYERS * DIM;     // 96*256    = 24576
  float* part     = enc + BATCH * NLAYERS * D4;           // 4*96*1024 = 393216

  // 1) scores: 25088 waves, 8 per block, float4 loads
  scores_kernel<<<(BATCH * SEQ) / 8, 256, 0, stream>>>(feat, pe, attn_w, scores);
  // 2) softmax in place over S
  softmax_kernel<<<BATCH * NLAYERS, 256, 0, stream>>>(scores);
  // 3a) partial attended per S-chunk (second feature sweep from L2)
  attend_part_kernel<<<BATCH * SCHUNK, 256, 0, stream>>>(feat, pe, scores, part);
  // 3b) deterministic chunk reduction
  attend_reduce_kernel<<<(BATCH * NLAYERS * DIM) / 256, 256, 0, stream>>>(part, attended);
  // 4) enc GEMM + bias + relu (WMMA f32), one wave per 16x16 tile
  enc_gemm_kernel<<<96, 32, 0, stream>>>(attended, enc_W, enc_b, enc);
  // 5) classifier GEMM + bias (WMMA f32)
  cls_gemm_kernel<<<50, 32, 0, stream>>>(enc, cls_W, cls_b, out);
}